// NMF2D_12730283065692
// MI455X (gfx1250) — compile-verified
//
#include <hip/hip_runtime.h>
#include <math.h>

#define D_ 512
#define N_ 4096
#define R_ 64
#define B_ 16
#define EPS_ 1e-6f

#define KCH_ 128    // K rows per LDS chunk (64 K-pairs)
#define PR_ 160     // dwords per K-pair row: pair-interleaved cols (128) + 32 pad
                    // -> consecutive pair-rows land on disjoint 32-bank halves

typedef __attribute__((ext_vector_type(2))) float v2f;
typedef __attribute__((ext_vector_type(8))) float v8f;

static __device__ __forceinline__ v8f wmma_f32(v2f a, v2f b, v8f c) {
  // V_WMMA_F32_16X16X4_F32: D = A(16x4) * B(4x16) + C(16x16), all fp32
  return __builtin_amdgcn_wmma_f32_16x16x4_f32(false, a, false, b, (short)0, c,
                                               false, false);
}

static __device__ __forceinline__ v2f ld2(const float* p) {
  return *(const v2f*)p;  // callers guarantee 8-byte alignment
}

// Cooperative fill of one K-chunk of a [K,64] row-major panel into LDS with
// K-pair interleave: (k, r) -> sP[(k>>1)*PR_ + 2r + (k&1)].
// Each B-operand fetch then becomes one contiguous ds_load_b64 (no reg movs).
// Tasks: 64 pair-rows x 16 col-segments (4 cols) = 1024 -> 4 per thread.
static __device__ __forceinline__ void fill_pair_panel(float* sP,
                                                       const float* gsrc,
                                                       int tid) {
#pragma unroll
  for (int i = 0; i < ((KCH_ / 2) * 16) / 256; ++i) {
    const int task = tid + i * 256;
    const int pr = task >> 4;
    const int seg = task & 15;
    const float4 a = *(const float4*)(gsrc + (size_t)(2 * pr) * R_ + seg * 4);
    const float4 c = *(const float4*)(gsrc + (size_t)(2 * pr + 1) * R_ + seg * 4);
    float* dst = sP + pr * PR_ + seg * 8;
    v2f t0; t0.x = a.x; t0.y = c.x; *(v2f*)(dst + 0) = t0;
    v2f t1; t1.x = a.y; t1.y = c.y; *(v2f*)(dst + 2) = t1;
    v2f t2; t2.x = a.z; t2.y = c.z; *(v2f*)(dst + 4) = t2;
    v2f t3; t3.x = a.w; t3.y = c.w; *(v2f*)(dst + 6) = t3;
  }
}

// numer[b,n,r] = sum_d x[b,d,n] * basW[b,d,r]   (N x R, K = D)
// wave computes a 16x64 strip; B panel (bases) staged pair-interleaved in LDS
__global__ void __launch_bounds__(256) k_numer(const float* __restrict__ x,
                                               const float* __restrict__ basW,
                                               float* __restrict__ numer) {
  __shared__ float sP[(KCH_ / 2) * PR_];  // 40 KB
  const int tid = threadIdx.x;
  const int lane = tid & 31;
  const int gwave = blockIdx.x * 8 + (tid >> 5);
  const int b = gwave >> 8;            // 256 strips per batch, 8 waves/block
  const int n0 = (gwave & 255) << 4;
  const float* xb = x + (size_t)b * D_ * N_;
  const float* Bb = basW + (size_t)b * D_ * R_;
  float* Ob = numer + (size_t)b * N_ * R_;
  const int m = lane & 15;
  const int hi = lane >> 4;
  const int kb = hi << 1;
  v8f acc0 = {}, acc1 = {}, acc2 = {}, acc3 = {};
  for (int c0 = 0; c0 < D_; c0 += KCH_) {
    __syncthreads();
    fill_pair_panel(sP, Bb + (size_t)c0 * R_, tid);
    __syncthreads();
    for (int kq = 0; kq < KCH_; kq += 4) {
      const float* xr = xb + (size_t)(c0 + kq + kb) * N_ + n0 + m;  // A = x^T
      v2f a;
      a.x = xr[0];
      a.y = xr[N_];
      const float* sp = sP + ((kq >> 1) + hi) * PR_;  // pair-row for this lane
      v2f b0 = ld2(sp + 2 * m);
      v2f b1 = ld2(sp + 2 * (16 + m));
      v2f b2 = ld2(sp + 2 * (32 + m));
      v2f b3 = ld2(sp + 2 * (48 + m));
      acc0 = wmma_f32(a, b0, acc0);
      acc1 = wmma_f32(a, b1, acc1);
      acc2 = wmma_f32(a, b2, acc2);
      acc3 = wmma_f32(a, b3, acc3);
    }
  }
#pragma unroll
  for (int v = 0; v < 8; ++v) {
    float* orow = Ob + (size_t)(n0 + v + 8 * hi) * R_ + m;
    orow[0]  = acc0[v];
    orow[16] = acc1[v];
    orow[32] = acc2[v];
    orow[48] = acc3[v];
  }
}

// numer2[b,d,r] = sum_n x[b,d,n] * coef[b,n,r]   (D x R, K = N)
__global__ void __launch_bounds__(256) k_numer2(const float* __restrict__ x,
                                                const float* __restrict__ coef,
                                                float* __restrict__ numer2) {
  __shared__ float sP[(KCH_ / 2) * PR_];  // 40 KB
  const int tid = threadIdx.x;
  const int lane = tid & 31;
  const int gwave = blockIdx.x * 8 + (tid >> 5);
  const int b = gwave >> 5;            // 32 strips per batch, 8 waves/block
  const int d0 = (gwave & 31) << 4;
  const float* xb = x + (size_t)b * D_ * N_;
  const float* Cb = coef + (size_t)b * N_ * R_;
  float* Ob = numer2 + (size_t)b * D_ * R_;
  const int m = lane & 15;
  const int hi = lane >> 4;
  const int kb = hi << 1;
  v8f acc0 = {}, acc1 = {}, acc2 = {}, acc3 = {};
  for (int c0 = 0; c0 < N_; c0 += KCH_) {
    __syncthreads();
    fill_pair_panel(sP, Cb + (size_t)c0 * R_, tid);
    __syncthreads();
    for (int kq = 0; kq < KCH_; kq += 4) {
      v2f a = ld2(xb + (size_t)(d0 + m) * N_ + c0 + kq + kb);  // A = x rows
      const float* sp = sP + ((kq >> 1) + hi) * PR_;
      v2f b0 = ld2(sp + 2 * m);
      v2f b1 = ld2(sp + 2 * (16 + m));
      v2f b2 = ld2(sp + 2 * (32 + m));
      v2f b3 = ld2(sp + 2 * (48 + m));
      acc0 = wmma_f32(a, b0, acc0);
      acc1 = wmma_f32(a, b1, acc1);
      acc2 = wmma_f32(a, b2, acc2);
      acc3 = wmma_f32(a, b3, acc3);
    }
  }
#pragma unroll
  for (int v = 0; v < 8; ++v) {
    float* orow = Ob + (size_t)(d0 + v + 8 * hi) * R_ + m;
    orow[0]  = acc0[v];
    orow[16] = acc1[v];
    orow[32] = acc2[v];
    orow[48] = acc3[v];
  }
}

// coef[row,:] = softmax(logits[row,:]) over R
__global__ void __launch_bounds__(256) k_softmax(const float* __restrict__ logits,
                                                 float* __restrict__ coef) {
  const int row = blockIdx.x * blockDim.x + threadIdx.x;  // grid sized exactly
  const float* p = logits + (size_t)row * R_;
  float* o = coef + (size_t)row * R_;
  float mx = -3.4e38f;
  for (int r = 0; r < R_; ++r) mx = fmaxf(mx, p[r]);
  float s = 0.f;
  for (int r = 0; r < R_; ++r) s += __expf(p[r] - mx);
  const float inv = 1.f / s;
  for (int r = 0; r < R_; ++r) o[r] = __expf(p[r] - mx) * inv;
}

// Split-K Gram partials: part[b,c] = Ac^T Ac over chunk c of K rows.
// A = src[b] : [kChunk*nChunks, 64] row-major. One block (16 waves) per chunk.
__global__ void __launch_bounds__(512) k_gram_part(const float* __restrict__ src,
                                                   int kChunk, int nChunks,
                                                   float* __restrict__ part) {
  const int b = blockIdx.x / nChunks;
  const int c = blockIdx.x % nChunks;
  const float* Ab = src + ((size_t)b * nChunks + c) * kChunk * R_;
  float* Pb = part + ((size_t)b * nChunks + c) * R_ * R_;
  const int lane = threadIdx.x & 31;
  const int w = threadIdx.x >> 5;
  const int m0 = (w >> 2) << 4;
  const int r0 = (w & 3) << 4;
  const int m = lane & 15;
  const int hi = lane >> 4;
  const int kb = hi << 1;
  v8f acc = {};
  for (int k0 = 0; k0 < kChunk; k0 += 4) {
    const float* row0 = Ab + (size_t)(k0 + kb) * R_;
    const float* row1 = row0 + R_;
    v2f a;  a.x = row0[m0 + m];  a.y = row1[m0 + m];
    v2f bb; bb.x = row0[r0 + m]; bb.y = row1[r0 + m];
    acc = wmma_f32(a, bb, acc);
  }
#pragma unroll
  for (int v = 0; v < 8; ++v)
    Pb[(size_t)(m0 + v + 8 * hi) * R_ + r0 + m] = acc[v];
}

// G[b] = sum_c part[b,c]  (deterministic split-K reduction, no atomics)
__global__ void __launch_bounds__(256) k_gram_reduce(const float* __restrict__ part,
                                                     int nChunks,
                                                     float* __restrict__ G) {
  const int i = blockIdx.x * 256 + threadIdx.x;  // < B_*R_*R_ (grid exact)
  const int b = i / (R_ * R_);
  const int e = i % (R_ * R_);
  float s = 0.f;
  for (int c = 0; c < nChunks; ++c)
    s += part[((size_t)b * nChunks + c) * R_ * R_ + e];
  G[i] = s;
}

// Fused multiplicative update: M = M * numer / (M·G + eps), M:[rows, R] per batch.
// Block owns 16 rows; 4 waves each compute one 16x16 denom tile via WMMA (K=64),
// buffer old values, barrier (all reads done), then write in place.
__global__ void __launch_bounds__(128) k_scale_update(float* __restrict__ M,
                                                      const float* __restrict__ numer,
                                                      const float* __restrict__ G,
                                                      int rows) {
  const int blocksPerBatch = rows >> 4;
  const int b = blockIdx.x / blocksPerBatch;
  const int n0 = (blockIdx.x % blocksPerBatch) << 4;
  float* Mb = M + (size_t)b * rows * R_;
  const float* Nb = numer + (size_t)b * rows * R_;
  const float* Gb = G + (size_t)b * R_ * R_;
  const int lane = threadIdx.x & 31;
  const int r0 = (threadIdx.x >> 5) << 4;
  const int m = lane & 15;
  const int hi = lane >> 4;
  const int kb = hi << 1;
  v8f acc = {};
#pragma unroll
  for (int k0 = 0; k0 < R_; k0 += 4) {
    v2f a = ld2(Mb + (size_t)(n0 + m) * R_ + k0 + kb);  // A = M rows (old)
    const float* g0 = Gb + (size_t)(k0 + kb) * R_ + r0 + m;
    v2f bb; bb.x = g0[0]; bb.y = g0[R_];
    acc = wmma_f32(a, bb, acc);
  }
  float oldv[8], nm[8];
#pragma unroll
  for (int v = 0; v < 8; ++v) {
    const size_t idx = (size_t)(n0 + v + 8 * hi) * R_ + r0 + m;
    oldv[v] = Mb[idx];
    nm[v] = Nb[idx];
  }
  __syncthreads();  // all reads of M complete before any wave overwrites
#pragma unroll
  for (int v = 0; v < 8; ++v) {
    const size_t idx = (size_t)(n0 + v + 8 * hi) * R_ + r0 + m;
    Mb[idx] = oldv[v] * nm[v] / (acc[v] + EPS_);
  }
}

// out[b,d,n] = sum_r basW[b,d,r] * coef[b,n,r]   (D x N, K = R)
__global__ void __launch_bounds__(256) k_out(const float* __restrict__ basW,
                                             const float* __restrict__ coef,
                                             float* __restrict__ out) {
  const int lane = threadIdx.x & 31;
  const int gwave = blockIdx.x * 8 + (threadIdx.x >> 5);
  const int b = gwave >> 11;           // 2048 waves per batch
  const int rem = gwave & 2047;
  const int d0 = (rem >> 6) << 4;      // 32 row strips
  const int n0 = (rem & 63) << 6;      // 64 col tiles of width 64
  const float* Ab = basW + (size_t)b * D_ * R_;
  const float* Cb = coef + (size_t)b * N_ * R_;
  float* Ob = out + (size_t)b * D_ * N_;
  const int m = lane & 15;
  const int hi = lane >> 4;
  const int kb = hi << 1;
  v8f acc0 = {}, acc1 = {}, acc2 = {}, acc3 = {};
#pragma unroll
  for (int k0 = 0; k0 < R_; k0 += 4) {
    v2f a  = ld2(Ab + (size_t)(d0 + m) * R_ + k0 + kb);        // A = bases
    v2f b0 = ld2(Cb + (size_t)(n0 + m) * R_ + k0 + kb);        // B = coef^T
    v2f b1 = ld2(Cb + (size_t)(n0 + 16 + m) * R_ + k0 + kb);
    v2f b2 = ld2(Cb + (size_t)(n0 + 32 + m) * R_ + k0 + kb);
    v2f b3 = ld2(Cb + (size_t)(n0 + 48 + m) * R_ + k0 + kb);
    acc0 = wmma_f32(a, b0, acc0);
    acc1 = wmma_f32(a, b1, acc1);
    acc2 = wmma_f32(a, b2, acc2);
    acc3 = wmma_f32(a, b3, acc3);
  }
#pragma unroll
  for (int v = 0; v < 8; ++v) {
    float* orow = Ob + (size_t)(d0 + v + 8 * hi) * N_ + n0 + m;
    orow[0]  = acc0[v];
    orow[16] = acc1[v];
    orow[32] = acc2[v];
    orow[48] = acc3[v];
  }
}

extern "C" void kernel_launch(void* const* d_in, const int* in_sizes, int n_in,
                              void* d_out, int out_size, void* d_ws, size_t ws_size,
                              hipStream_t stream) {
  (void)in_sizes; (void)n_in; (void)out_size; (void)ws_size;
  const float* x = (const float*)d_in[0];      // [16, 512, 64, 64]
  const float* bases = (const float*)d_in[1];  // [16, 512, 64]
  float* out = (float*)d_out;                  // [16, 512, 64, 64]

  float* ws = (float*)d_ws;
  float* coef  = ws;                                  // 16*4096*64  = 16 MB
  float* tmpNR = coef  + (size_t)B_ * N_ * R_;        // 16*4096*64  = 16 MB
  float* basW  = tmpNR + (size_t)B_ * N_ * R_;        // 16*512*64   =  2 MB
  float* tmpDR = basW  + (size_t)B_ * D_ * R_;        // 16*512*64   =  2 MB
  float* G     = tmpDR + (size_t)B_ * D_ * R_;        // 16*64*64    = 256 KB
  float* part  = G     + (size_t)B_ * R_ * R_;        // 16*8*64*64  =  2 MB

  const int gramRed = (B_ * R_ * R_) / 256;           // 256 blocks

  // working copy of bases (inputs must not be mutated)
  hipMemcpyAsync(basW, bases, (size_t)B_ * D_ * R_ * sizeof(float),
                 hipMemcpyDeviceToDevice, stream);

  // logits = x^T . bases ; coef = softmax(logits). These logits equal the
  // first iteration's numer (bases unchanged), so reuse tmpNR.
  k_numer<<<512, 256, 0, stream>>>(x, basW, tmpNR);
  k_softmax<<<(B_ * N_) / 256, 256, 0, stream>>>(tmpNR, coef);

  for (int it = 0; it < 7; ++it) {
    // BtB (K=512, 2 chunks of 256)
    k_gram_part<<<B_ * 2, 512, 0, stream>>>(basW, 256, 2, part);
    k_gram_reduce<<<gramRed, 256, 0, stream>>>(part, 2, G);
    k_scale_update<<<B_ * (N_ / 16), 128, 0, stream>>>(coef, tmpNR, G, N_);
    k_numer2<<<64, 256, 0, stream>>>(x, coef, tmpDR);
    // CtC (K=4096, 8 chunks of 512)
    k_gram_part<<<B_ * 8, 512, 0, stream>>>(coef, 512, 8, part);
    k_gram_reduce<<<gramRed, 256, 0, stream>>>(part, 8, G);
    k_scale_update<<<B_ * (D_ / 16), 128, 0, stream>>>(basW, tmpDR, G, D_);
    k_numer<<<512, 256, 0, stream>>>(x, basW, tmpNR);  // numer for next update
  }

  // compute_coef: one final coef update with fresh numer (already in tmpNR)
  k_gram_part<<<B_ * 2, 512, 0, stream>>>(basW, 256, 2, part);
  k_gram_reduce<<<gramRed, 256, 0, stream>>>(part, 2, G);
  k_scale_update<<<B_ * (N_ / 16), 128, 0, stream>>>(coef, tmpNR, G, N_);

  // out = bases . coef^T
  k_out<<<4096, 256, 0, stream>>>(basW, coef, out);
}